// CharEncoder_17695265259952
// MI455X (gfx1250) — compile-verified
//
#include <hip/hip_runtime.h>
#include <hip/hip_bf16.h>

typedef float v2f __attribute__((ext_vector_type(2)));
typedef float v8f __attribute__((ext_vector_type(8)));

static constexpr int kVocab = 256;
static constexpr int kE     = 64;
static constexpr int kH     = 256;
static constexpr int kW     = 20;     // chars per word
static constexpr int kWords = 64 * 128;
static constexpr int kF3    = 3 * kH; // 768

// ---------------------------------------------------------------------------
// Kernel 1: per-tap lookup tables  T[tap][c][h] = sum_e emb[c][e]*w[h][e][j]
// (+ bias baked into each branch's center tap).  9 taps, each a 256x256x64
// f32 GEMM done with V_WMMA_F32_16X16X4_F32.  One wave per 16x16 D tile.
// ---------------------------------------------------------------------------
__global__ __launch_bounds__(32) void build_tables(
    const float* __restrict__ emb,
    const float* __restrict__ w1, const float* __restrict__ b1,
    const float* __restrict__ w3, const float* __restrict__ b3,
    const float* __restrict__ w5, const float* __restrict__ b5,
    float* __restrict__ T)
{
    int tile  = blockIdx.x;          // 9 * 16 * 16 tiles
    int tap   = tile >> 8;           // 0..8
    int cTile = (tile >> 4) & 15;    // vocab tile (M)
    int hTile = tile & 15;           // h tile (N)

    const float* wp; const float* bp; int kk, j;
    if (tap == 0)     { wp = w1; bp = b1; kk = 1; j = 0; }
    else if (tap < 4) { wp = w3; bp = b3; kk = 3; j = tap - 1; }
    else              { wp = w5; bp = b5; kk = 5; j = tap - 4; }
    const bool center = (j == (kk >> 1));

    const int lane = threadIdx.x & 31;
    const int half = lane >> 4;      // 0 or 1 -> K pair {0,1} vs {2,3}
    const int l16  = lane & 15;

    const int row = cTile * 16 + l16;   // A row  (M = vocab char)
    const int col = hTile * 16 + l16;   // B col  (N = h)

    v8f acc = {};
    for (int e0 = 0; e0 < kE; e0 += 4) {
        int ea = e0 + 2 * half;
        // A[M,K] from emb[row][ea..ea+1]  (contiguous -> float2)
        v2f a = *(const v2f*)(emb + row * kE + ea);
        // B[K,N] from w[col][e][j], stride kk over e
        v2f b;
        b.x = wp[(col * kE + ea    ) * kk + j];
        b.y = wp[(col * kE + ea + 1) * kk + j];
        acc = __builtin_amdgcn_wmma_f32_16x16x4_f32(
            false, a, false, b, (short)0, acc, false, false);
    }

    const float bias = center ? bp[col] : 0.0f;
    float* outp = T + tap * (kVocab * kH) + (cTile * 16) * kH + hTile * 16;
#pragma unroll
    for (int i = 0; i < 8; ++i) {
        int m = half * 8 + i;            // D layout: M = 8*half + vgpr
        outp[m * kH + l16] = acc[i] + bias;
    }
}

// ---------------------------------------------------------------------------
// Kernel 2: conv-as-lookup + relu + global max over positions.
// One block per word; thread tid handles column h=tid of all 3 branches.
// Loads are lane-contiguous in h -> fully coalesced 128B/wave.
// ---------------------------------------------------------------------------
__global__ __launch_bounds__(256) void conv_gather(
    const int*   __restrict__ chars,   // [kWords][kW]
    const float* __restrict__ T,       // [9][256][256]
    float*       __restrict__ feat)    // [kWords][768]
{
    __shared__ int c[kW];
    const int n   = blockIdx.x;
    const int tid = threadIdx.x;
    if (tid < kW) c[tid] = chars[n * kW + tid];
    __syncthreads();

    const int h = tid;
    const float* T0 = T;                  // k=1 center (bias baked)
    const float* T1 = T + 1 * 65536;      // k=3: c[t-1]
    const float* T2 = T + 2 * 65536;      // k=3: c[t]   (bias baked)
    const float* T3 = T + 3 * 65536;      // k=3: c[t+1]
    const float* T4 = T + 4 * 65536;      // k=5: c[t-2]
    const float* T5 = T + 5 * 65536;      // k=5: c[t-1]
    const float* T6 = T + 6 * 65536;      // k=5: c[t]   (bias baked)
    const float* T7 = T + 7 * 65536;      // k=5: c[t+1]
    const float* T8 = T + 8 * 65536;      // k=5: c[t+2]

    float m1 = 0.0f, m3 = 0.0f, m5 = 0.0f;   // relu>=0, so 0 is valid init
#pragma unroll
    for (int t = 0; t < kW; ++t) {
        float s1 = T0[c[t] * kH + h];
        m1 = fmaxf(m1, s1);

        float s3 = T2[c[t] * kH + h];
        if (t >= 1)      s3 += T1[c[t - 1] * kH + h];
        if (t <= kW - 2) s3 += T3[c[t + 1] * kH + h];
        m3 = fmaxf(m3, s3);

        float s5 = T6[c[t] * kH + h];
        if (t >= 2)      s5 += T4[c[t - 2] * kH + h];
        if (t >= 1)      s5 += T5[c[t - 1] * kH + h];
        if (t <= kW - 2) s5 += T7[c[t + 1] * kH + h];
        if (t <= kW - 3) s5 += T8[c[t + 2] * kH + h];
        m5 = fmaxf(m5, s5);
    }

    feat[n * kF3 + h]          = m1;
    feat[n * kF3 + kH + h]     = m3;
    feat[n * kF3 + 2 * kH + h] = m5;
}

// ---------------------------------------------------------------------------
// Kernel 3: out[8192,256] = feat[8192,768] @ lw^T + lb  via f32 WMMA 16x16x4.
// One 16x16 tile per wave32, K loop of 192; A and B K-pairs via float2 loads.
// ---------------------------------------------------------------------------
__global__ __launch_bounds__(256) void linear_wmma(
    const float* __restrict__ feat,   // [kWords][768]
    const float* __restrict__ lw,     // [256][768]
    const float* __restrict__ lb,     // [256]
    float*       __restrict__ out)    // [kWords][256]
{
    const int wave  = blockIdx.x * (blockDim.x >> 5) + (threadIdx.x >> 5);
    const int mTile = wave >> 4;      // 0..511
    const int nTile = wave & 15;      // 0..15
    const int lane  = threadIdx.x & 31;
    const int half  = lane >> 4;
    const int l16   = lane & 15;

    const float* arow = feat + (mTile * 16 + l16) * kF3 + 2 * half;
    const float* brow = lw   + (nTile * 16 + l16) * kF3 + 2 * half;

    v8f acc = {};
#pragma unroll 4
    for (int k0 = 0; k0 < kF3; k0 += 4) {
        v2f a = *(const v2f*)(arow + k0);
        v2f b = *(const v2f*)(brow + k0);
        acc = __builtin_amdgcn_wmma_f32_16x16x4_f32(
            false, a, false, b, (short)0, acc, false, false);
    }

    const float bias = lb[nTile * 16 + l16];   // D column per lane = l16
    float* o = out + (mTile * 16) * kH + nTile * 16;
#pragma unroll
    for (int i = 0; i < 8; ++i) {
        int m = half * 8 + i;
        o[m * kH + l16] = acc[i] + bias;
    }
}

// ---------------------------------------------------------------------------
extern "C" void kernel_launch(void* const* d_in, const int* in_sizes, int n_in,
                              void* d_out, int out_size, void* d_ws, size_t ws_size,
                              hipStream_t stream)
{
    const int*   chars = (const int*)  d_in[0];
    const float* emb   = (const float*)d_in[1];
    const float* w1    = (const float*)d_in[2];
    const float* b1    = (const float*)d_in[3];
    const float* w3    = (const float*)d_in[4];
    const float* b3    = (const float*)d_in[5];
    const float* w5    = (const float*)d_in[6];
    const float* b5    = (const float*)d_in[7];
    const float* lw    = (const float*)d_in[8];
    const float* lb    = (const float*)d_in[9];
    float*       outp  = (float*)d_out;

    float* T    = (float*)d_ws;                  // 9*256*256 f32 = 2.25 MB
    float* feat = T + 9 * kVocab * kH;           // 8192*768  f32 = 24 MB

    // 9 taps * 16 * 16 tiles, one wave each
    build_tables<<<9 * 16 * 16, 32, 0, stream>>>(emb, w1, b1, w3, b3, w5, b5, T);

    // one block per word
    conv_gather<<<kWords, 256, 0, stream>>>(chars, T, feat);

    // 512 * 16 = 8192 tile-waves, 8 waves per block
    linear_wmma<<<(kWords / 16) * (kH / 16) / 8, 256, 0, stream>>>(feat, lw, lb, outp);
}